// GIN_3layer_basic_71949292143004
// MI455X (gfx1250) — compile-verified
//
#include <hip/hip_runtime.h>
#include <stdint.h>

// ---------------------------------------------------------------------------
// GIN 3-layer forward for gfx1250 (MI455X).
//   layer(x) = relu_opt( (x + scatter_sum(x[src] -> dst)) @ W + b )
// - Aggregation: native f32 global atomics (unsafeAtomicAdd ->
//   GLOBAL_ATOMIC_ADD_F32, no CAS loop) into an L2-resident accumulator that
//   is pre-initialized to x (fuses the "+x" self term). ~256M atomics
//   dominate the roofline; all hot buffers fit in the 192MB L2.
// - GEMM: V_WMMA_F32_16X16X4_F32 (exact fp32), fully unrolled via
//   compile-time shapes; weights repacked once into K-pair-major float2
//   layout so every B fragment is one contiguous global_load_b64.
// ---------------------------------------------------------------------------

typedef __attribute__((ext_vector_type(2))) float v2f;
typedef __attribute__((ext_vector_type(8))) float v8f;

#define GIN_N 100000
#define GIN_E 1600000

// ---------------------------------------------------------------- utilities
__global__ void k_copy_f4(const float4* __restrict__ in,
                          float4* __restrict__ out, int n4) {
  int i = blockIdx.x * blockDim.x + threadIdx.x;
  if (i < n4) out[i] = in[i];
}

// Narrow int64 edge indices to int32 (read 3x by scatters, written once).
__global__ void k_idx32(const long long* __restrict__ src,
                        const long long* __restrict__ dst,
                        int* __restrict__ s32, int* __restrict__ d32, int E) {
  int i = blockIdx.x * blockDim.x + threadIdx.x;
  if (i < E) {
    s32[i] = (int)src[i];
    d32[i] = (int)dst[i];
  }
}

// Repack W[K, Nsrc] (row-major) into Wp[p, col] = {W[2p][col], W[2p+1][col]},
// col padded with zeros up to Np. Also pads bias to Np.
__global__ void k_repack_w(const float* __restrict__ W, const float* __restrict__ b,
                           float2* __restrict__ Wp, float* __restrict__ bp,
                           int Khalf, int Nsrc, int Np) {
  int tid = blockIdx.x * blockDim.x + threadIdx.x;
  if (tid < Np) bp[tid] = (tid < Nsrc) ? b[tid] : 0.0f;
  int total = Khalf * Np;
  if (tid >= total) return;
  int p = tid / Np;
  int col = tid - p * Np;
  float2 v;
  if (col < Nsrc) {
    v.x = W[(size_t)(2 * p) * Nsrc + col];
    v.y = W[(size_t)(2 * p + 1) * Nsrc + col];
  } else {
    v.x = 0.0f; v.y = 0.0f;
  }
  Wp[tid] = v;
}

// ------------------------------------------------------------------ scatter
// One thread per (edge, feature-pair). hlog2 = log2(d/2). Within a wave all
// lanes of the same edge share the index load; gathers/atomics coalesced.
// unsafeAtomicAdd guarantees GLOBAL_ATOMIC_ADD_F32 (no-return, STOREcnt).
__global__ void k_scatter_add(const int* __restrict__ src,
                              const int* __restrict__ dst,
                              const float2* __restrict__ h2,
                              float* __restrict__ acc,
                              int total, int hlog2, int hmask) {
  int tid = blockIdx.x * blockDim.x + threadIdx.x;
  if (tid >= total) return;
  int e  = tid >> hlog2;
  int f2 = tid & hmask;
  int s = src[e];
  int t = dst[e];
  float2 v = h2[(((size_t)s) << hlog2) + f2];
  float* p = acc + ((((size_t)t) << hlog2) << 1) + (f2 << 1);
  unsafeAtomicAdd(p, v.x);
  unsafeAtomicAdd(p + 1, v.y);
}

// --------------------------------------------------------------------- GEMM
// out[N, NOUT_STORE] = relu_opt(A[N, K] @ W + bias), W pre-packed (see above).
// One wave per 16-row tile; N % 16 == 0 and grid exact -> EXEC all-1s.
// f32 WMMA layouts (ISA 7.12.2):
//   A 16x4:  row m=lane&15; lanes<16 hold {K=k0,k0+1}, lanes>=16 {k0+2,k0+3}
//   B 4x16:  col n=lane&15; same K split by half-wave
//   C 16x16: VGPR g -> row g + 8*(lane>>4), col n
template <int K, int NOUTP, int NOUT_STORE, int RELU>
__global__ void k_gin_gemm_wmma(const float* __restrict__ A,
                                const float2* __restrict__ Wp,
                                const float* __restrict__ bp,
                                float* __restrict__ out) {
  const int lane    = threadIdx.x & 31;
  const int wave    = threadIdx.x >> 5;
  const int rowTile = blockIdx.x * (blockDim.x >> 5) + wave;
  const int base    = rowTile << 4;
  const int n       = lane & 15;
  const int kh      = lane >> 4;

  // A fragments for this 16-row tile: one aligned float2 per K-step.
  const float2* __restrict__ arow2 =
      (const float2*)(A + (size_t)(base + n) * K);
  v2f afrag[K / 4];
#pragma unroll
  for (int i = 0; i < K / 4; ++i) {
    float2 t = arow2[2 * i + kh];   // ka = 4i + 2*kh
    afrag[i].x = t.x;
    afrag[i].y = t.y;
  }

#pragma unroll
  for (int t = 0; t < NOUTP / 16; ++t) {
    const int col = (t << 4) + n;
    v8f c = {};
#pragma unroll
    for (int i = 0; i < K / 4; ++i) {
      float2 bw = Wp[(size_t)(2 * i + kh) * NOUTP + col];
      v2f b; b.x = bw.x; b.y = bw.y;
      c = __builtin_amdgcn_wmma_f32_16x16x4_f32(
          false, afrag[i], false, b, (short)0, c, false, false);
    }
    if (NOUT_STORE == NOUTP || col < NOUT_STORE) {
      const float bv = bp[col];
#pragma unroll
      for (int g = 0; g < 8; ++g) {
        const int row = base + (kh << 3) + g;
        float v = c[g] + bv;
        if (RELU) v = fmaxf(v, 0.0f);
        out[(size_t)row * NOUT_STORE + col] = v;
      }
    }
  }
}

// ------------------------------------------------------------------- launch
extern "C" void kernel_launch(void* const* d_in, const int* in_sizes, int n_in,
                              void* d_out, int out_size, void* d_ws, size_t ws_size,
                              hipStream_t stream) {
  const float*     x  = (const float*)d_in[0];
  const long long* ei = (const long long*)d_in[1];   // int64 [2, E]
  const float*     W1 = (const float*)d_in[2];
  const float*     b1 = (const float*)d_in[3];
  const float*     W2 = (const float*)d_in[4];
  const float*     b2 = (const float*)d_in[5];
  const float*     W3 = (const float*)d_in[6];
  const float*     b3 = (const float*)d_in[7];

  // ---- workspace layout (all 8B-aligned by construction) ----
  float*  acc = (float*)d_ws;                          // N*64 f32
  float*  h   = acc + (size_t)GIN_N * 64;              // N*64 f32
  int*    s32 = (int*)(h + (size_t)GIN_N * 64);        // E i32
  int*    d32 = s32 + GIN_E;                           // E i32
  float2* wp1 = (float2*)(d32 + GIN_E);                // 16*64   float2
  float2* wp2 = wp1 + (32 / 2) * 64;                   // 32*64   float2
  float2* wp3 = wp2 + (64 / 2) * 64;                   // 32*16   float2
  float*  bp1 = (float*)(wp3 + (64 / 2) * 16);         // 64 f32
  float*  bp2 = bp1 + 64;                              // 64 f32
  float*  bp3 = bp2 + 64;                              // 16 f32

  const dim3 gemmBlock(64);                 // 2 waves / block
  const dim3 gemmGrid(GIN_N / 16 / 2);      // 3125 (exact, no guards)

  // ---- one-shot prep: narrow indices, repack weights ----
  k_idx32<<<(GIN_E + 255) / 256, 256, 0, stream>>>(ei, ei + GIN_E, s32, d32, GIN_E);
  k_repack_w<<<(16 * 64 + 255) / 256, 256, 0, stream>>>(W1, b1, wp1, bp1, 16, 64, 64);
  k_repack_w<<<(32 * 64 + 255) / 256, 256, 0, stream>>>(W2, b2, wp2, bp2, 32, 64, 64);
  k_repack_w<<<(32 * 16 + 255) / 256, 256, 0, stream>>>(W3, b3, wp3, bp3, 32, 10, 16);

  // ---- Layer 1: 32 -> 64, relu ----
  {
    int n4 = GIN_N * 32 / 4;
    k_copy_f4<<<(n4 + 255) / 256, 256, 0, stream>>>((const float4*)x, (float4*)acc, n4);
    int total = GIN_E * 16;   // E * d/2
    k_scatter_add<<<(total + 255) / 256, 256, 0, stream>>>(
        s32, d32, (const float2*)x, acc, total, 4, 15);
    k_gin_gemm_wmma<32, 64, 64, 1><<<gemmGrid, gemmBlock, 0, stream>>>(acc, wp1, bp1, h);
  }
  // ---- Layer 2: 64 -> 64, relu ----
  {
    int n4 = GIN_N * 64 / 4;
    k_copy_f4<<<(n4 + 255) / 256, 256, 0, stream>>>((const float4*)h, (float4*)acc, n4);
    int total = GIN_E * 32;
    k_scatter_add<<<(total + 255) / 256, 256, 0, stream>>>(
        s32, d32, (const float2*)h, acc, total, 5, 31);
    k_gin_gemm_wmma<64, 64, 64, 1><<<gemmGrid, gemmBlock, 0, stream>>>(acc, wp2, bp2, h);
  }
  // ---- Layer 3: 64 -> 10 (padded to 16), no relu ----
  {
    int n4 = GIN_N * 64 / 4;
    k_copy_f4<<<(n4 + 255) / 256, 256, 0, stream>>>((const float4*)h, (float4*)acc, n4);
    int total = GIN_E * 32;
    k_scatter_add<<<(total + 255) / 256, 256, 0, stream>>>(
        s32, d32, (const float2*)h, acc, total, 5, 31);
    k_gin_gemm_wmma<64, 16, 10, 0><<<gemmGrid, gemmBlock, 0, stream>>>(acc, wp3, bp3, (float*)d_out);
  }
}